// MultiHeadAttention_34136400068718
// MI455X (gfx1250) — compile-verified
//
#include <hip/hip_runtime.h>
#include <math.h>

typedef __attribute__((ext_vector_type(2))) float v2f;
typedef __attribute__((ext_vector_type(8))) float v8f;

#define D_MODEL 768
#define NUM_HEADS 12
#define D_HEAD 64
#define SEQ 2048
#define BATCH 2
#define ROWS (BATCH * SEQ)              // 4096
#define QKV_ELEMS (BATCH * NUM_HEADS * SEQ * D_HEAD)  // 3,145,728
#define OUT_ELEMS (ROWS * D_MODEL)      // 3,145,728

// ---------------------------------------------------------------------------
// Generic wave-level GEMM tile: C[16 x 64] += A[16 x K] * B, f32 WMMA 16x16x4.
//   BT==true : B is stored as [N,K] row-major and used transposed (X @ W^T).
//   BT==false: B is stored as [K,N] row-major (plain A @ B).
//   SM==0 : plain row-major store into C (ldc), + optional bias, * scale
//   SM==1 : head-split store: row r -> (b = r/2048, s = r%2048),
//           col c -> (h = c/64, d = c%64), OUT[b][h][s][d]   (for Q/K/V)
// One wave computes a 16x64 C tile (4 accumulators) reusing the A fragment.
// wid must be wave-uniform; EXEC is all ones at every WMMA.
// ---------------------------------------------------------------------------
template <bool BT, int SM>
__device__ __forceinline__ void gemm_tile(const float* __restrict__ A,
                                          const float* __restrict__ B,
                                          const float* __restrict__ bias,
                                          float* __restrict__ C,
                                          int M, int N, int K,
                                          int lda, int ldb, int ldc,
                                          float scale, int wid) {
    const int mtiles = M >> 4;
    const int mt = wid % mtiles;
    const int nb = wid / mtiles;            // 64-wide N block
    if (nb >= (N >> 6)) return;             // wave-uniform guard

    const int lane = threadIdx.x & 31;
    const int lh   = lane >> 4;             // half-wave select (K group / row+8)
    const int lm   = lane & 15;

    const int arow = mt * 16 + lm;
    const float* arowp = A + (size_t)arow * lda;

    v8f acc0 = {}, acc1 = {}, acc2 = {}, acc3 = {};
    const int n0 = nb * 64;

    for (int k = 0; k < K; k += 4) {
        const int ks = k + 2 * lh;          // this half-wave's K pair
        v2f a = *(const v2f*)(arowp + ks);  // A[row][ks], A[row][ks+1]

        v2f b0, b1, b2, b3;
        if (BT) {
            b0 = *(const v2f*)(B + (size_t)(n0 + 0 * 16 + lm) * ldb + ks);
            b1 = *(const v2f*)(B + (size_t)(n0 + 1 * 16 + lm) * ldb + ks);
            b2 = *(const v2f*)(B + (size_t)(n0 + 2 * 16 + lm) * ldb + ks);
            b3 = *(const v2f*)(B + (size_t)(n0 + 3 * 16 + lm) * ldb + ks);
        } else {
            const float* r0 = B + (size_t)ks * ldb + n0 + lm;
            const float* r1 = r0 + ldb;
            b0.x = r0[0];  b0.y = r1[0];
            b1.x = r0[16]; b1.y = r1[16];
            b2.x = r0[32]; b2.y = r1[32];
            b3.x = r0[48]; b3.y = r1[48];
        }
        acc0 = __builtin_amdgcn_wmma_f32_16x16x4_f32(false, a, false, b0, (short)0, acc0, false, false);
        acc1 = __builtin_amdgcn_wmma_f32_16x16x4_f32(false, a, false, b1, (short)0, acc1, false, false);
        acc2 = __builtin_amdgcn_wmma_f32_16x16x4_f32(false, a, false, b2, (short)0, acc2, false, false);
        acc3 = __builtin_amdgcn_wmma_f32_16x16x4_f32(false, a, false, b3, (short)0, acc3, false, false);
    }

    v8f accs[4] = {acc0, acc1, acc2, acc3};
#pragma unroll
    for (int t = 0; t < 4; ++t) {
        const int col = n0 + t * 16 + lm;
        const float bv = bias ? bias[col] : 0.0f;
#pragma unroll
        for (int i = 0; i < 8; ++i) {
            const int r = mt * 16 + i + 8 * lh;
            const float val = accs[t][i] * scale + bv;
            if (SM == 0) {
                C[(size_t)r * ldc + col] = val;
            } else {
                const int bI = r >> 11, s = r & 2047;
                const int h = col >> 6, d = col & 63;
                C[((((size_t)bI * NUM_HEADS + h) * SEQ) + s) * D_HEAD + d] = val;
            }
        }
    }
}

// --- QKV projection: OUT[b,h,s,d] = X @ W^T + bias ------------------------
__global__ void __launch_bounds__(256)
k_proj_split(const float* __restrict__ X, const float* __restrict__ W,
             const float* __restrict__ bias, float* __restrict__ OUT) {
    const int wid = (blockIdx.x * 256 + threadIdx.x) >> 5;
    gemm_tile<true, 1>(X, W, bias, OUT, ROWS, D_MODEL, D_MODEL,
                       D_MODEL, D_MODEL, 0, 1.0f, wid);
}

// --- scores = Q_bh @ K_bh^T / 8 into attn region --------------------------
__global__ void __launch_bounds__(256)
k_scores(const float* __restrict__ Q, const float* __restrict__ Kb,
         float* __restrict__ attn) {
    const int bh = blockIdx.y;
    const float* A = Q  + (size_t)bh * SEQ * D_HEAD;
    const float* B = Kb + (size_t)bh * SEQ * D_HEAD;
    float* C = attn + (size_t)bh * SEQ * SEQ;
    const int wid = (blockIdx.x * 256 + threadIdx.x) >> 5;
    gemm_tile<true, 0>(A, B, nullptr, C, SEQ, SEQ, D_HEAD,
                       D_HEAD, D_HEAD, SEQ, 0.125f, wid);
}

// --- row softmax in place over attn [B*H*S rows of length SEQ] ------------
__global__ void __launch_bounds__(256)
k_softmax(float* __restrict__ attn) {
    float* p = attn + (size_t)blockIdx.x * SEQ;
    __shared__ float red[256];
    const int t = threadIdx.x;

    float v[8];
    float mx = -INFINITY;
#pragma unroll
    for (int i = 0; i < 8; ++i) {
        v[i] = p[t + i * 256];
        mx = fmaxf(mx, v[i]);
    }
    red[t] = mx;
    __syncthreads();
    for (int s = 128; s > 0; s >>= 1) {
        if (t < s) red[t] = fmaxf(red[t], red[t + s]);
        __syncthreads();
    }
    mx = red[0];
    __syncthreads();

    float sum = 0.0f;
#pragma unroll
    for (int i = 0; i < 8; ++i) {
        v[i] = __expf(v[i] - mx);
        sum += v[i];
    }
    red[t] = sum;
    __syncthreads();
    for (int s = 128; s > 0; s >>= 1) {
        if (t < s) red[t] += red[t + s];
        __syncthreads();
    }
    const float inv = 1.0f / red[0];
#pragma unroll
    for (int i = 0; i < 8; ++i) p[t + i * 256] = v[i] * inv;
}

// --- heads = attn_bh @ V_bh, merged-head store into H [4096 x 768] --------
__global__ void __launch_bounds__(256)
k_attnv(const float* __restrict__ attn, const float* __restrict__ V,
        float* __restrict__ Hbuf) {
    const int bh = blockIdx.y;
    const int bI = bh / NUM_HEADS, h = bh % NUM_HEADS;
    const float* A = attn + (size_t)bh * SEQ * SEQ;
    const float* B = V + (size_t)bh * SEQ * D_HEAD;
    float* C = Hbuf + (size_t)bI * SEQ * D_MODEL + h * D_HEAD;  // ldc = D_MODEL
    const int wid = (blockIdx.x * 256 + threadIdx.x) >> 5;
    gemm_tile<false, 0>(A, B, nullptr, C, SEQ, D_HEAD, SEQ,
                        SEQ, D_HEAD, D_MODEL, 1.0f, wid);
}

// --- out = H @ W_o^T + b_o ------------------------------------------------
__global__ void __launch_bounds__(256)
k_outproj(const float* __restrict__ Hbuf, const float* __restrict__ W,
          const float* __restrict__ bias, float* __restrict__ OUT) {
    const int wid = (blockIdx.x * 256 + threadIdx.x) >> 5;
    gemm_tile<true, 0>(Hbuf, W, bias, OUT, ROWS, D_MODEL, D_MODEL,
                       D_MODEL, D_MODEL, D_MODEL, 1.0f, wid);
}

extern "C" void kernel_launch(void* const* d_in, const int* in_sizes, int n_in,
                              void* d_out, int out_size, void* d_ws, size_t ws_size,
                              hipStream_t stream) {
    const float* query = (const float*)d_in[0];
    const float* key   = (const float*)d_in[1];
    const float* value = (const float*)d_in[2];
    const float* W_q   = (const float*)d_in[3];
    const float* b_q   = (const float*)d_in[4];
    const float* W_k   = (const float*)d_in[5];
    const float* b_k   = (const float*)d_in[6];
    const float* W_v   = (const float*)d_in[7];
    const float* b_v   = (const float*)d_in[8];
    const float* W_o   = (const float*)d_in[9];
    const float* b_o   = (const float*)d_in[10];

    float* out  = (float*)d_out;                 // [B,S,D_MODEL]
    float* attn = out + OUT_ELEMS;               // [B,H,S,S]

    float* Q  = (float*)d_ws;                    // [B,H,S,Dh]
    float* Kb = Q  + QKV_ELEMS;
    float* V  = Kb + QKV_ELEMS;
    float* Hb = V  + QKV_ELEMS;                  // [ROWS, D_MODEL]

    // QKV projections: 256 M-tiles * 12 N64-blocks = 3072 waves -> 384 blocks
    k_proj_split<<<dim3(384), 256, 0, stream>>>(query, W_q, b_q, Q);
    k_proj_split<<<dim3(384), 256, 0, stream>>>(key,   W_k, b_k, Kb);
    k_proj_split<<<dim3(384), 256, 0, stream>>>(value, W_v, b_v, V);

    // scores: per (b,h) 128 M-tiles * 32 N64-blocks = 4096 waves -> 512 blocks
    k_scores<<<dim3(512, BATCH * NUM_HEADS), 256, 0, stream>>>(Q, Kb, attn);

    // softmax: one block per row, B*H*S rows
    k_softmax<<<dim3(BATCH * NUM_HEADS * SEQ), 256, 0, stream>>>(attn);

    // attn @ V: per (b,h) 128 M-tiles * 1 N64-block = 128 waves -> 16 blocks
    k_attnv<<<dim3(16, BATCH * NUM_HEADS), 256, 0, stream>>>(attn, V, Hb);

    // final projection
    k_outproj<<<dim3(384), 256, 0, stream>>>(Hb, W_o, b_o, out);
}